// HighQualityNSN2NLoss_73744588472448
// MI455X (gfx1250) — compile-verified
//
#include <hip/hip_runtime.h>

// ---------------- problem constants ----------------
#define H_IMG 768
#define W_IMG 768
#define B_IMG 16
#define HW    (H_IMG*W_IMG)
#define N_TOT (B_IMG*HW)            // 9,437,184  (< 2^24, counts exact in f32)
#define TILE  32
#define RAD3  9                     // gauss sigma=3 -> 19 taps
#define RAW   (TILE + 2*RAD3)       // 50
#define RAWS  52                    // padded LDS row stride
#define H3R   52                    // 50 rows + 2 zero pad rows (K padded to 36)
#define H1R   40                    // 38 rows + 2 zero pad rows (K padded to 24)
#define MIN_BODY 4096

typedef float v2f __attribute__((ext_vector_type(2)));
typedef float v8f __attribute__((ext_vector_type(8)));

#if defined(__has_builtin)
#  if __has_builtin(__builtin_amdgcn_wmma_f32_16x16x4_f32)
#    define USE_WMMA 1
#  endif
#endif
#ifndef USE_WMMA
#  define USE_WMMA 0
#endif

// monotone float<->uint key for radix select
__device__ __forceinline__ unsigned fkey(float f) {
  unsigned u = __float_as_uint(f);
  return u ^ ((u & 0x80000000u) ? 0xFFFFFFFFu : 0x80000000u);
}
__device__ __forceinline__ float keyToFloat(unsigned k) {
  unsigned u = (k & 0x80000000u) ? (k ^ 0x80000000u) : ~k;
  return __uint_as_float(u);
}

// Vertical banded-conv of a horizontally pre-blurred LDS tile via WMMA f32 16x16x4.
// out[m][n] = sum_d g[d] * hbuf[(qy+m+d)*TILE + qx+n]   (m,n in 0..15)
// As matmul: A[m][k] = g[k-m] (banded, K padded to 4*CHUNKS), B[k][n] = hbuf row qy+k.
// Assumed f32 layouts (half-wave K split, consistent with ISA A-16x4 table):
//   A/B chunk c: lanes 0-15 hold K=4c,4c+1 in VGPR0,1 ; lanes 16-31 hold K=4c+2,4c+3.
//   D: VGPR r -> (M=r, N=lane) for lanes 0-15 ; (M=r+8, N=lane-16) for lanes 16-31.
template <int TAPS, int CHUNKS>
__device__ __forceinline__ v8f vconv(const float* hbuf, const v2f* afrag,
                                     const float* g, int qy, int qx, int lane) {
#if USE_WMMA
  const int  MN   = lane & 15;               // A row / B col
  const int  koff = (lane >= 16) ? 2 : 0;
  v2f bfrag[CHUNKS];
  #pragma unroll
  for (int c = 0; c < CHUNKS; ++c) {         // all B loads first -> one dscnt wait
    const int k0 = 4 * c + koff;
    bfrag[c].x = hbuf[(qy + k0    ) * TILE + qx + MN];
    bfrag[c].y = hbuf[(qy + k0 + 1) * TILE + qx + MN];
  }
  v8f acc = {};
  #pragma unroll
  for (int c = 0; c < CHUNKS; ++c)
    acc = __builtin_amdgcn_wmma_f32_16x16x4_f32(false, afrag[c], false, bfrag[c],
                                                (short)0, acc, false, false);
  return acc;
#else
  v8f out = {};
  const int n   = lane & 15;
  const int off = (lane >= 16) ? 8 : 0;
  #pragma unroll
  for (int r = 0; r < 8; ++r) {
    float s = 0.0f;
    #pragma unroll
    for (int d = 0; d < TAPS; ++d)
      s += g[d] * hbuf[(qy + off + r + d) * TILE + qx + n];
    out[r] = s;
  }
  return out;
#endif
}

// ---------------- kernel A: fused conv + masked reductions + coarse hist ----------------
__global__ __launch_bounds__(128)
void loss_main(const float* __restrict__ yp,  const float* __restrict__ npred,
               const float* __restrict__ xi,  const float* __restrict__ xm,
               const float* __restrict__ wt,  const float* __restrict__ nsyn,
               float* __restrict__ acc,
               unsigned* __restrict__ histYP, unsigned* __restrict__ histXI) {
  __shared__ float sG3[19], sG1[7];
  __shared__ float raw[3][RAW * RAWS];   // 0:x_i 1:y_pred 2:x_mid (halo 9, zero-padded)
  __shared__ float h3[3][H3R * TILE];    // horizontal 19-tap results (+2 zero rows)
  __shared__ float h1[2][H1R * TILE];    // horizontal 7-tap results (+2 zero rows)

  const int tid = threadIdx.x;
  const int b   = blockIdx.z;
  const int X0  = blockIdx.x * TILE, Y0 = blockIdx.y * TILE;
  const size_t planeOff = (size_t)b * HW + (size_t)Y0 * W_IMG + X0;

  // Prefetch the phase-4 streamed arrays: one prefetch per tile row; each tile
  // row is exactly one 128B cacheline (32 floats, 128B aligned since X0%32==0).
  {
    int rowsel = tid & 31, which = tid >> 5;   // 4 groups of 32 threads
    const float* pf = (which == 0) ? wt : (which == 1) ? npred : (which == 2) ? nsyn : nullptr;
    if (pf) __builtin_prefetch(pf + planeOff + (size_t)rowsel * W_IMG, 0, 3);
  }

  if (tid == 0) {
    float s3 = 0.f, s1 = 0.f;
    for (int j = 0; j < 19; ++j) { float c = (float)(j - 9) / 3.0f; float e = expf(-0.5f * c * c); sG3[j] = e; s3 += e; }
    for (int j = 0; j < 19; ++j) sG3[j] /= s3;
    for (int j = 0; j < 7;  ++j) { float c = (float)(j - 3);        float e = expf(-0.5f * c * c); sG1[j] = e; s1 += e; }
    for (int j = 0; j < 7;  ++j) sG1[j] /= s1;
  }

  const float* pXI = xi + (size_t)b * HW;
  const float* pYP = yp + (size_t)b * HW;
  const float* pXM = xm + (size_t)b * HW;

  for (int idx = tid; idx < RAW * RAW; idx += 128) {
    int ry = idx / RAW, rx = idx - ry * RAW;
    int gy = Y0 - RAD3 + ry, gx = X0 - RAD3 + rx;
    bool in = (gy >= 0) & (gy < H_IMG) & (gx >= 0) & (gx < W_IMG);
    int gi = gy * W_IMG + gx;
    raw[0][ry * RAWS + rx] = in ? pXI[gi] : 0.0f;
    raw[1][ry * RAWS + rx] = in ? pYP[gi] : 0.0f;
    raw[2][ry * RAWS + rx] = in ? pXM[gi] : 0.0f;
  }
  __syncthreads();

  // Per-lane banded A-matrix fragments from the LDS weight tables.
  // Unconditional clamped-index loads (LDS broadcast) + one compare-select:
  // no exec-mask predication, no expf/div chains. (d==clamp(d) <=> in band.)
  const int lane = tid & 31;
  const int MN   = lane & 15;
  const int koff = (lane >= 16) ? 2 : 0;
  v2f a3[9], a1[6];
  #pragma unroll
  for (int c = 0; c < 9; ++c) {
    int d0 = 4 * c + koff - MN, d1 = d0 + 1;
    int i0 = min(max(d0, 0), 18), i1 = min(max(d1, 0), 18);
    float w0 = sG3[i0], w1 = sG3[i1];
    a3[c].x = (d0 == i0) ? w0 : 0.0f;
    a3[c].y = (d1 == i1) ? w1 : 0.0f;
  }
  #pragma unroll
  for (int c = 0; c < 6; ++c) {
    int d0 = 4 * c + koff - MN, d1 = d0 + 1;
    int i0 = min(max(d0, 0), 6), i1 = min(max(d1, 0), 6);
    float w0 = sG1[i0], w1 = sG1[i1];
    a1[c].x = (d0 == i0) ? w0 : 0.0f;
    a1[c].y = (d1 == i1) ? w1 : 0.0f;
  }

  // horizontal passes (VALU)
  for (int idx = tid; idx < H3R * TILE; idx += 128) {
    int ry = idx >> 5, cx = idx & 31;
    float a0 = 0.f, a1v = 0.f, a2 = 0.f;
    if (ry < RAW) {
      #pragma unroll
      for (int j = 0; j < 19; ++j) {
        float g = sG3[j];
        a0  += g * raw[0][ry * RAWS + cx + j];
        a1v += g * raw[1][ry * RAWS + cx + j];
        a2  += g * raw[2][ry * RAWS + cx + j];
      }
    }
    h3[0][idx] = a0; h3[1][idx] = a1v; h3[2][idx] = a2;
  }
  for (int idx = tid; idx < H1R * TILE; idx += 128) {
    int ry = idx >> 5, cx = idx & 31;
    float a0 = 0.f, a1v = 0.f;
    if (ry < 38) {
      #pragma unroll
      for (int j = 0; j < 7; ++j) {
        float g = sG1[j];
        a0  += g * raw[0][(ry + 6) * RAWS + cx + 6 + j];
        a1v += g * raw[1][(ry + 6) * RAWS + cx + 6 + j];
      }
    }
    h1[0][idx] = a0; h1[1][idx] = a1v;
  }
  __syncthreads();

  const int wv = tid >> 5;
  const int qy = (wv >> 1) * 16, qx = (wv & 1) * 16;

  // vertical passes on the matrix pipe
  v8f lowI = vconv<19, 9>(h3[0], a3, sG3, qy, qx, lane);
  v8f lowP = vconv<19, 9>(h3[1], a3, sG3, qy, qx, lane);
  v8f lowM = vconv<19, 9>(h3[2], a3, sG3, qy, qx, lane);
  v8f g1I  = vconv< 7, 6>(h1[0], a1, sG1, qy, qx, lane);
  v8f g1P  = vconv< 7, 6>(h1[1], a1, sG1, qy, qx, lane);

  float a_rc = 0, a_cb = 0, a_sxi = 0, a_syp = 0, a_edge = 0, a_ct = 0, a_stex = 0;
  float a_cf = 0, a_cfb = 0, a_lf = 0, a_mid = 0, a_hf = 0, a_syn = 0, a_ic = 0;

  const int n    = lane & 15;
  const int moff = (lane >= 16) ? 8 : 0;

  #pragma unroll
  for (int r = 0; r < 8; ++r) {
    const int m  = moff + r;
    const int ly = qy + m, lx = qx + n;
    const int cy = ly + RAD3, cx9 = lx + RAD3;
    #define R0(dy, dx) raw[0][(cy + (dy)) * RAWS + cx9 + (dx)]
    #define R1(dy, dx) raw[1][(cy + (dy)) * RAWS + cx9 + (dx)]
    float c_xi = R0(0, 0), c_yp = R1(0, 0);
    float gxi = (R0(-1, 1) + 2.f * R0(0, 1) + R0(1, 1)) - (R0(-1, -1) + 2.f * R0(0, -1) + R0(1, -1));
    float gyi = (R0(1, -1) + 2.f * R0(1, 0) + R0(1, 1)) - (R0(-1, -1) + 2.f * R0(-1, 0) + R0(-1, 1));
    float gxp = (R1(-1, 1) + 2.f * R1(0, 1) + R1(1, 1)) - (R1(-1, -1) + 2.f * R1(0, -1) + R1(1, -1));
    float gyp = (R1(1, -1) + 2.f * R1(1, 0) + R1(1, 1)) - (R1(-1, -1) + 2.f * R1(-1, 0) + R1(-1, 1));
    float lap_i = R0(-1, 0) + R0(1, 0) + R0(0, -1) + R0(0, 1) - 4.f * R0(0, 0);
    float lap_p = R1(-1, 0) + R1(1, 0) + R1(0, -1) + R1(0, 1) - 4.f * R1(0, 0);
    #undef R0
    #undef R1
    const size_t gidx = (size_t)b * HW + (size_t)(Y0 + ly) * W_IMG + (X0 + lx);
    float wv_ = wt[gidx], np_ = npred[gidx], ns_ = nsyn[gidx];

    a_rc += fabsf(c_yp * wv_ - c_xi * wv_);
    float gmi = sqrtf(gxi * gxi + gyi * gyi + 1e-8f);
    float gmp = sqrtf(gxp * gxp + gyp * gyp + 1e-8f);
    a_edge += fabsf(gxp - gxi) + fabsf(gyp - gyi);
    bool body = (c_xi > 0.15f) && (c_xi < 0.85f);
    bool flat = gmi < 0.03f;
    if (gmi > 0.03f && gmi < 0.5f) { a_ct += 1.f; a_stex += fabsf(gmp - gmi); }
    if (flat) {
      a_cf += 1.f;
      a_hf += fabsf(fabsf(lap_p) - 0.3f * fabsf(lap_i));
      a_ic += fmaxf(gmp - 2.0f * gmi, 0.0f);
    }
    if (body) {
      a_cb += 1.f; a_sxi += c_xi; a_syp += c_yp;
      atomicAdd(&histYP[fkey(c_yp) >> 16], 1u);
      atomicAdd(&histXI[fkey(c_xi) >> 16], 1u);
      if (flat) {
        a_cfb += 1.f;
        a_lf  += fabsf((lowP[r] - lowM[r]) - 0.3f * (lowI[r] - lowM[r]));
        float mi = g1I[r] - lowI[r], mp = g1P[r] - lowP[r];
        a_mid += fabsf(fabsf(mp) - 0.3f * fabsf(mi));
        a_syn += fabsf(np_ - ns_);
      }
    }
  }

  #define WRED(x, i) { float _v = (x); \
    for (int _o = 16; _o > 0; _o >>= 1) _v += __shfl_down(_v, _o, 32); \
    if (lane == 0) atomicAdd(&acc[i], _v); }
  WRED(a_rc, 0)  WRED(a_cb, 1)  WRED(a_sxi, 2)  WRED(a_syp, 3)
  WRED(a_edge, 4) WRED(a_ct, 5) WRED(a_stex, 6) WRED(a_cf, 7)
  WRED(a_cfb, 8) WRED(a_lf, 9)  WRED(a_mid, 10) WRED(a_hf, 11)
  WRED(a_syn, 12) WRED(a_ic, 13)
  #undef WRED
}

// ---------------- kernel B: coarse histogram scan -> (bin, residual rank) per query ----------------
__global__ void scan_coarse(const float* __restrict__ acc,
                            const unsigned* __restrict__ histYP,
                            const unsigned* __restrict__ histXI,
                            unsigned* __restrict__ qmeta) {
  __shared__ unsigned pre[1024];
  const int t = blockIdx.x, tid = threadIdx.x;
  const unsigned* hist = (t == 0) ? histYP : histXI;
  const int nb = (int)acc[1];
  if (nb <= MIN_BODY) {           // loss_hu gated off; mark queries as skip
    if (tid < 4) { qmeta[t * 4 + tid] = 0xFFFFFFFFu; qmeta[8 + t * 4 + tid] = 0u; }
    return;
  }
  const float nm1 = (float)(nb - 1);
  float p25 = 0.25f * nm1, p75 = 0.75f * nm1;
  int lo25 = (int)floorf(p25), hi25 = (int)ceilf(p25);
  int lo75 = (int)floorf(p75), hi75 = (int)ceilf(p75);
  lo25 = max(0, min(lo25, N_TOT - 1)); hi25 = max(0, min(hi25, N_TOT - 1));
  lo75 = max(0, min(lo75, N_TOT - 1)); hi75 = max(0, min(hi75, N_TOT - 1));
  unsigned ranks[4] = { (unsigned)lo25, (unsigned)hi25, (unsigned)lo75, (unsigned)hi75 };

  const int base = tid * 64;
  unsigned part = 0;
  for (int j = 0; j < 64; ++j) part += hist[base + j];
  pre[tid] = part;
  __syncthreads();
  if (tid == 0) { unsigned run = 0; for (int i = 0; i < 1024; ++i) { unsigned p = pre[i]; pre[i] = run; run += p; } }
  __syncthreads();
  unsigned cum = pre[tid];
  for (int j = 0; j < 64; ++j) {
    unsigned c = hist[base + j];
    if (c) {
      for (int qi = 0; qi < 4; ++qi) {
        unsigned r = ranks[qi];
        if (r >= cum && r < cum + c) { qmeta[t * 4 + qi] = (unsigned)(base + j); qmeta[8 + t * 4 + qi] = r - cum; }
      }
    }
    cum += c;
  }
}

// ---------------- kernel C: refined (low 16-bit) histograms for matching coarse bins ----------------
__global__ void refine_hist(const float* __restrict__ yp, const float* __restrict__ xi,
                            const unsigned* __restrict__ qmeta, unsigned* __restrict__ refined) {
  const unsigned qb0 = qmeta[0], qb1 = qmeta[1], qb2 = qmeta[2], qb3 = qmeta[3];
  const unsigned qb4 = qmeta[4], qb5 = qmeta[5], qb6 = qmeta[6], qb7 = qmeta[7];
  int i = blockIdx.x * blockDim.x + threadIdx.x;
  const int stride = gridDim.x * blockDim.x;
  for (; i < N_TOT; i += stride) {
    float x = xi[i];
    if (x > 0.15f && x < 0.85f) {
      unsigned k = fkey(yp[i]); unsigned cb = k >> 16, lw = k & 0xFFFFu;
      if (cb == qb0) atomicAdd(&refined[0 * 65536 + lw], 1u);
      if (cb == qb1) atomicAdd(&refined[1 * 65536 + lw], 1u);
      if (cb == qb2) atomicAdd(&refined[2 * 65536 + lw], 1u);
      if (cb == qb3) atomicAdd(&refined[3 * 65536 + lw], 1u);
      unsigned k2 = fkey(x); unsigned cb2 = k2 >> 16, lw2 = k2 & 0xFFFFu;
      if (cb2 == qb4) atomicAdd(&refined[4 * 65536 + lw2], 1u);
      if (cb2 == qb5) atomicAdd(&refined[5 * 65536 + lw2], 1u);
      if (cb2 == qb6) atomicAdd(&refined[6 * 65536 + lw2], 1u);
      if (cb2 == qb7) atomicAdd(&refined[7 * 65536 + lw2], 1u);
    }
  }
}

// ---------------- kernel D: refined scan -> exact order-statistic values ----------------
__global__ void scan_refined(const unsigned* __restrict__ qmeta,
                             const unsigned* __restrict__ refined,
                             float* __restrict__ qval) {
  __shared__ unsigned pre[1024];
  const int q = blockIdx.x, tid = threadIdx.x;
  const unsigned bin = qmeta[q];
  if (bin == 0xFFFFFFFFu) { if (tid == 0) qval[q] = 0.0f; return; }
  const unsigned* h = refined + (size_t)q * 65536;
  const unsigned rank = qmeta[8 + q];
  const int base = tid * 64;
  unsigned part = 0;
  for (int j = 0; j < 64; ++j) part += h[base + j];
  pre[tid] = part;
  __syncthreads();
  if (tid == 0) { unsigned run = 0; for (int i = 0; i < 1024; ++i) { unsigned p = pre[i]; pre[i] = run; run += p; } }
  __syncthreads();
  unsigned cum = pre[tid];
  for (int j = 0; j < 64; ++j) {
    unsigned c = h[base + j];
    if (c && rank >= cum && rank < cum + c) {
      unsigned key = (bin << 16) | (unsigned)(base + j);
      qval[q] = keyToFloat(key);
    }
    cum += c;
  }
}

// ---------------- kernel E: scalar combine ----------------
__global__ void finalize(const float* __restrict__ acc, const float* __restrict__ qval,
                         float* __restrict__ out) {
  const float Nf = (float)N_TOT;
  float rc = acc[0] / Nf;
  float nbf = acc[1]; int nb = (int)nbf;
  float den_b = fmaxf(nbf, 1.0f);
  float mean_in = acc[2] / den_b, mean_pr = acc[3] / den_b;
  int nn = nb > 1 ? nb : 1;
  float nm1 = (float)(nn - 1);
  float p25 = 0.25f * nm1, p75 = 0.75f * nm1;
  float f25 = p25 - floorf(p25), f75 = p75 - floorf(p75);
  float q_yp25 = qval[0] * (1.f - f25) + qval[1] * f25;
  float q_yp75 = qval[2] * (1.f - f75) + qval[3] * f75;
  float q_xi25 = qval[4] * (1.f - f25) + qval[5] * f25;
  float q_xi75 = qval[6] * (1.f - f75) + qval[7] * f75;
  float dmu = mean_pr - mean_in, d25 = q_yp25 - q_xi25, d75 = q_yp75 - q_xi75;
  float hu = dmu * dmu + 0.5f * (d25 * d25 + d75 * d75);
  float loss_hu = (nb > MIN_BODY) ? hu : 0.0f;
  float edge = acc[4] / Nf;
  float ct = acc[5];
  float tex = (ct > 100.0f) ? acc[6] / fmaxf(ct, 1.0f) : 0.0f;
  float nf = acc[7], nfb = acc[8];
  float lf  = (nfb > 100.0f) ? acc[9]  / fmaxf(nfb, 1.0f) : 0.0f;
  float mid = (nfb > 100.0f) ? acc[10] / fmaxf(nfb, 1.0f) : 0.0f;
  float hf  = (nf  > 100.0f) ? acc[11] / fmaxf(nf,  1.0f) : 0.0f;
  float syn = (nfb > 100.0f) ? acc[12] / fmaxf(nfb, 1.0f) : 0.0f;
  float ic  = (nf  > 100.0f) ? acc[13] / fmaxf(nf,  1.0f) : 0.0f;
  out[0] = 2.0f * rc + 1.5f * loss_hu + 1.0f * edge + 0.8f * tex +
           1.5f * hf + 0.8f * mid + 0.6f * lf + 1.0f * syn + 0.8f * ic;
}

// ---------------- launch ----------------
extern "C" void kernel_launch(void* const* d_in, const int* in_sizes, int n_in,
                              void* d_out, int out_size, void* d_ws, size_t ws_size,
                              hipStream_t stream) {
  const float* yp    = (const float*)d_in[0];   // y_pred          (B,1,1,H,W)
  const float* npred = (const float*)d_in[1];   // noise_pred      (B,1,1,H,W)
  const float* xi    = (const float*)d_in[2];   // x_i             (B,1,H,W)
  // d_in[3] = x_ip1 : unused by the reference
  const float* xm    = (const float*)d_in[4];   // x_mid
  const float* wt    = (const float*)d_in[5];   // W
  const float* ns    = (const float*)d_in[6];   // noise_synthetic

  float*    acc    = (float*)d_ws;                                  // [0,64) floats
  unsigned* qmeta  = (unsigned*)((char*)d_ws + 256);                // qbin[8], qrank[8]
  float*    qval   = (float*)((char*)d_ws + 320);                   // 8 floats
  unsigned* histYP = (unsigned*)((char*)d_ws + 512);                // 65536 bins
  unsigned* histXI = histYP + 65536;                                // 65536 bins
  unsigned* refined = histXI + 65536;                               // 8 * 65536 bins

  const size_t WS_BYTES = 512 + (size_t)10 * 65536 * 4;             // ~2.5 MB
  hipMemsetAsync(d_ws, 0, WS_BYTES, stream);

  loss_main  <<<dim3(W_IMG / TILE, H_IMG / TILE, B_IMG), 128, 0, stream>>>
             (yp, npred, xi, xm, wt, ns, acc, histYP, histXI);
  scan_coarse<<<2, 1024, 0, stream>>>(acc, histYP, histXI, qmeta);
  refine_hist<<<4608, 256, 0, stream>>>(yp, xi, qmeta, refined);
  scan_refined<<<8, 1024, 0, stream>>>(qmeta, refined, qval);
  finalize   <<<1, 1, 0, stream>>>(acc, qval, (float*)d_out);
}